// DGCNNEncoderGn_13469017440293
// MI455X (gfx1250) — compile-verified
//
#include <hip/hip_runtime.h>
#include <hip/hip_bf16.h>

typedef __attribute__((ext_vector_type(2))) float v2f;
typedef __attribute__((ext_vector_type(8))) float v8f;

#define NPT   2048
#define KNN   80
#define BATCH 4
#define LEAK  0.2f
#define GEPS  1e-5f

// ---------------- float atomic max/min (sign-split int trick) ----------------
__device__ __forceinline__ void atomicMaxF(float* a, float v){
  if (v >= 0.f) atomicMax((int*)a, __float_as_int(v));
  else          atomicMin((unsigned int*)a, __float_as_uint(v));
}
__device__ __forceinline__ void atomicMinF(float* a, float v){
  if (v >= 0.f) atomicMin((int*)a, __float_as_int(v));
  else          atomicMax((unsigned int*)a, __float_as_uint(v));
}

// ---------------- xx[b,n] = sum_c x[b,c,n]^2 ----------------
__global__ void sqnorm_kernel(const float* __restrict__ x, long bStride, int C,
                              float* __restrict__ xx){
  int i = blockIdx.x*blockDim.x + threadIdx.x;       // b*NPT + n
  int b = i >> 11, n = i & (NPT-1);
  const float* p = x + (long)b*bStride + n;
  float s = 0.f;
  for (int c = 0; c < C; ++c){ float v = p[(long)c*NPT]; s += v*v; }
  xx[i] = s;
}

// ---------------- KNN top-80 of neg_dist, one workgroup per (b,n) ----------------
__global__ void knn_kernel(const float* __restrict__ x, long bStride, int C,
                           const float* __restrict__ xx, int* __restrict__ idxK){
  __shared__ float dist[NPT];
  __shared__ float ctr[64];
  __shared__ float rv[128];
  __shared__ int   ri[128];
  const int tid = threadIdx.x;
  const int n = blockIdx.x, b = blockIdx.y;
  const float* xb = x + (long)b*bStride;
  if (tid < C) ctr[tid] = xb[(long)tid*NPT + n];
  __syncthreads();
  const float xxn = xx[b*NPT + n];
  for (int j = tid; j < NPT; j += 128){
    float dot = 0.f;
    for (int c = 0; c < C; ++c) dot += ctr[c]*xb[(long)c*NPT + j];
    dist[j] = 2.f*dot - xxn - xx[b*NPT + j];
  }
  __syncthreads();
  int* orow = idxK + ((long)(b*NPT + n))*KNN;
  for (int t = 0; t < KNN; ++t){
    float best = -__builtin_huge_valf(); int bi = NPT;
    for (int j = tid; j < NPT; j += 128){
      float d = dist[j];
      if (d > best || (d == best && j < bi)){ best = d; bi = j; }
    }
    rv[tid] = best; ri[tid] = bi;
    __syncthreads();
    for (int s = 64; s > 0; s >>= 1){
      if (tid < s){
        if (rv[tid+s] > rv[tid] || (rv[tid+s] == rv[tid] && ri[tid+s] < ri[tid])){
          rv[tid] = rv[tid+s]; ri[tid] = ri[tid+s];
        }
      }
      __syncthreads();
    }
    if (tid == 0){ orow[t] = ri[0]; dist[ri[0]] = -__builtin_huge_valf(); }
    __syncthreads();
  }
}

// ---------------- small init kernels ----------------
__global__ void zero_kernel(float* p, int n){
  int i = blockIdx.x*blockDim.x + threadIdx.x; if (i < n) p[i] = 0.f;
}
__global__ void initmm_kernel(float* hmx, float* hmn, int n){
  int i = blockIdx.x*blockDim.x + threadIdx.x;
  if (i < n){ hmx[i] = -__builtin_huge_valf(); hmn[i] = __builtin_huge_valf(); }
}

// ---------------- edge-conv pass 1: WMMA GEMM + fused stats (templated) ----------------
// h[o, n, k] = sum_c W[o,c] * F[c, n, k]; emits per-(b,o,n) max/min over k,
// and per-(b,group) sum/sumsq via atomics. One workgroup per (b, n). 8 waves.
template<int CIN, int COUT>
__global__ __launch_bounds__(256)
void edgeconv_pass1(const float* __restrict__ xin, long bStride,
                    const float* __restrict__ W, const int* __restrict__ idxK,
                    float* __restrict__ hmax, float* __restrict__ hmin,
                    float* __restrict__ gsum, float* __restrict__ gsumsq){
  constexpr int CF     = 2*CIN;
  constexpr int CFP    = (CF + 31) & ~31;
  constexpr int MTILES = COUT >> 4;

  __shared__ float Fs[32*KNN];   // 32-channel chunk of graph feature, 10 KB
  __shared__ float ctrS[CIN];    // center feature of this point
  __shared__ int   jdx[KNN];
  __shared__ float ls1[2], ls2[2];

  const int tid  = threadIdx.x;
  const int wid  = tid >> 5;
  const int lane = tid & 31;
  const int half = lane >> 4;
  const int l16  = lane & 15;
  const int koff = half << 1;          // VGPR pair holds K = {koff, koff+1}
  const int n = blockIdx.x, b = blockIdx.y;

  const float* xb = xin + (long)b*bStride;
  if (tid < KNN) jdx[tid] = idxK[((long)(b*NPT + n))*KNN + tid];
  if (tid < CIN) ctrS[tid] = xb[(long)tid*NPT + n];
  if (tid < 2){ ls1[tid] = 0.f; ls2[tid] = 0.f; }

  v8f acc[5] = {};                     // 5 pixel tiles (80 = 5x16) per wave
  const int mt = wid;                  // wave owns one 16-row tile of Cout
  const int m  = mt*16 + l16;          // A-matrix row supplied by this lane
  if (mt < MTILES) __builtin_prefetch(&W[(long)m*CF], 0, 1);  // global_prefetch_b8
  __syncthreads();

  for (int cc = 0; cc < CFP; cc += 32){
    // stage F[cc..cc+31][0..79]: c<CIN -> nbr-ctr, c<CF -> ctr, pad -> 0
    #pragma unroll
    for (int e0 = 0; e0 < 32*KNN; e0 += 256){
      int e = e0 + tid;
      int cl = e / KNN, k = e - cl*KNN;
      int c = cc + cl;
      float v = 0.f;
      if (c < CIN)      v = xb[(long)c*NPT + jdx[k]] - ctrS[c];
      else if (c < CF)  v = ctrS[c - CIN];
      Fs[e] = v;
    }
    __syncthreads();

    if (mt < MTILES){                  // wave-uniform: EXEC all-1s inside
      #pragma unroll
      for (int kk = 0; kk < 32; kk += 4){
        int c0 = cc + kk + koff;
        v2f a;
        a.x = (c0     < CF) ? W[(long)m*CF + c0    ] : 0.f;
        a.y = (c0 + 1 < CF) ? W[(long)m*CF + c0 + 1] : 0.f;
        #pragma unroll
        for (int pt = 0; pt < 5; ++pt){
          int pix = pt*16 + l16;
          v2f bf;
          bf.x = Fs[(kk + koff    )*KNN + pix];
          bf.y = Fs[(kk + koff + 1)*KNN + pix];
          acc[pt] = __builtin_amdgcn_wmma_f32_16x16x4_f32(
                        false, a, false, bf, (short)0, acc[pt], false, false);
        }
      }
    }
    __syncthreads();
  }

  if (mt < MTILES){
    float s1 = 0.f, s2 = 0.f;
    float mx[8], mn[8];
    #pragma unroll
    for (int v = 0; v < 8; ++v){ mx[v] = -__builtin_huge_valf(); mn[v] = __builtin_huge_valf(); }
    #pragma unroll
    for (int pt = 0; pt < 5; ++pt)
      #pragma unroll
      for (int v = 0; v < 8; ++v){
        float h = acc[pt][v];
        s1 += h; s2 += h*h;
        mx[v] = fmaxf(mx[v], h); mn[v] = fminf(mn[v], h);
      }
    for (int off = 16; off; off >>= 1){
      s1 += __shfl_xor(s1, off, 32);
      s2 += __shfl_xor(s2, off, 32);
    }
    #pragma unroll
    for (int v = 0; v < 8; ++v)
      for (int off = 8; off; off >>= 1){       // reduce within 16-lane half
        mx[v] = fmaxf(mx[v], __shfl_xor(mx[v], off, 32));
        mn[v] = fminf(mn[v], __shfl_xor(mn[v], off, 32));
      }
    if (l16 == 0){                             // lanes 0 and 16
      int mbase = mt*16 + half*8;              // C/D layout: M = v + 8*(lane>=16)
      #pragma unroll
      for (int v = 0; v < 8; ++v){
        long o = ((long)b*COUT + mbase + v)*NPT + n;
        hmax[o] = mx[v]; hmin[o] = mn[v];
      }
    }
    if (lane == 0){
      int g = (mt*16*2 >= COUT) ? 1 : 0;       // groups = 2
      atomicAdd(&ls1[g], s1); atomicAdd(&ls2[g], s2);
    }
  }
  __syncthreads();
  if (tid < 2){
    atomicAdd(&gsum  [b*2 + tid], ls1[tid]);
    atomicAdd(&gsumsq[b*2 + tid], ls2[tid]);
  }
}

// ---------------- edge-conv pass 2: affine + leaky via monotonicity ----------------
__global__ void ec_finalize(const float* __restrict__ hmax, const float* __restrict__ hmin,
                            const float* __restrict__ gsum, const float* __restrict__ gsumsq,
                            const float* __restrict__ gamma, const float* __restrict__ beta,
                            int Cout, float invcount,
                            float* __restrict__ out, long obStride){
  int i = blockIdx.x*blockDim.x + threadIdx.x;   // over B*Cout*NPT
  int n = i & (NPT-1);
  int t = i >> 11;
  int c = t % Cout, b = t / Cout;
  int g = (2*c >= Cout) ? 1 : 0;
  float mean = gsum[b*2+g]*invcount;
  float var  = gsumsq[b*2+g]*invcount - mean*mean;
  float a  = rsqrtf(var + GEPS)*gamma[c];
  float bb = beta[c] - mean*a;
  float v  = (a >= 0.f) ? hmax[i] : hmin[i];     // max_k leaky(a*h+b) = leaky(a*ext_k h + b)
  float y  = a*v + bb;
  y = (y >= 0.f) ? y : LEAK*y;
  out[(long)b*obStride + (long)c*NPT + n] = y;
}

// ---------------- final MLP pass 1: WMMA + bias + fused stats ----------------
__global__ __launch_bounds__(256)
void mlp_pass1(const float* __restrict__ feat, long bStride,
               const float* __restrict__ Wm, const float* __restrict__ bm,
               float* __restrict__ hmax2, float* __restrict__ hmin2,
               float* __restrict__ gs2, float* __restrict__ gss2){
  __shared__ float Ft[256*16];                  // full K=256 x 16-column tile, 16 KB
  __shared__ float gp1[8], gp2[8];
  const int tid = threadIdx.x, wid = tid >> 5, lane = tid & 31;
  const int half = lane >> 4, l16 = lane & 15, koff = half << 1;
  const int n0 = blockIdx.x*16, b = blockIdx.y;
  const float* fb = feat + (long)b*bStride;
  if (tid < 8){ gp1[tid] = 0.f; gp2[tid] = 0.f; }
  #pragma unroll
  for (int e0 = 0; e0 < 256*16; e0 += 256){
    int e = e0 + tid;
    int c = e >> 4, col = e & 15;
    Ft[e] = fb[(long)c*NPT + n0 + col];
  }
  __syncthreads();

  for (int mt = wid; mt < 64; mt += 8){         // 64 M-tiles of 1024 channels
    v8f acc = {};
    int m = mt*16 + l16;
    __builtin_prefetch(&Wm[(long)m*256], 0, 1);
    #pragma unroll 8
    for (int kk = 0; kk < 256; kk += 4){
      v2f a;
      a.x = Wm[(long)m*256 + kk + koff];
      a.y = Wm[(long)m*256 + kk + koff + 1];
      v2f bf;
      bf.x = Ft[(kk + koff    )*16 + l16];
      bf.y = Ft[(kk + koff + 1)*16 + l16];
      acc = __builtin_amdgcn_wmma_f32_16x16x4_f32(
                false, a, false, bf, (short)0, acc, false, false);
    }
    float s1 = 0.f, s2 = 0.f;
    int mbase = mt*16 + half*8;
    float mx[8], mn[8];
    #pragma unroll
    for (int v = 0; v < 8; ++v){
      float h = acc[v] + bm[mbase + v];
      s1 += h; s2 += h*h;
      mx[v] = h; mn[v] = h;
    }
    #pragma unroll
    for (int v = 0; v < 8; ++v)
      for (int off = 8; off; off >>= 1){
        mx[v] = fmaxf(mx[v], __shfl_xor(mx[v], off, 32));
        mn[v] = fminf(mn[v], __shfl_xor(mn[v], off, 32));
      }
    if (l16 == 0){
      #pragma unroll
      for (int v = 0; v < 8; ++v){
        atomicMaxF(&hmax2[b*1024 + mbase + v], mx[v]);
        atomicMinF(&hmin2[b*1024 + mbase + v], mn[v]);
      }
    }
    for (int off = 16; off; off >>= 1){ s1 += __shfl_xor(s1, off, 32); s2 += __shfl_xor(s2, off, 32); }
    if (lane == 0){ atomicAdd(&gp1[mt >> 3], s1); atomicAdd(&gp2[mt >> 3], s2); }
  }
  __syncthreads();
  if (tid < 8){ atomicAdd(&gs2[b*8 + tid], gp1[tid]); atomicAdd(&gss2[b*8 + tid], gp2[tid]); }
}

// ---------------- final MLP pass 2 ----------------
__global__ void mlp_finalize(const float* __restrict__ hmax2, const float* __restrict__ hmin2,
                             const float* __restrict__ gs2, const float* __restrict__ gss2,
                             const float* __restrict__ gm, const float* __restrict__ betam,
                             float* __restrict__ x4){
  int i = blockIdx.x*blockDim.x + threadIdx.x;  // 4096
  int b = i >> 10, o = i & 1023;
  int g = o >> 7;                               // groups = 8
  const float invc = 1.f/(128.f*2048.f);
  float mean = gs2[b*8+g]*invc;
  float var  = gss2[b*8+g]*invc - mean*mean;
  float a  = rsqrtf(var + GEPS)*gm[o];
  float bb = betam[o] - mean*a;
  float v  = (a >= 0.f) ? hmax2[i] : hmin2[i];
  x4[i] = fmaxf(a*v + bb, 0.f);
}

// ---------------- host orchestration ----------------
template<int CIN, int COUT>
static void run_edgeconv(const float* xin, long bStride,
                         const float* W, const float* gamma, const float* beta,
                         float* outbase, long obStride,
                         int* idxK, float* xx, float* hmax, float* hmin,
                         float* gsum, float* gsumsq, hipStream_t s){
  sqnorm_kernel<<<BATCH*NPT/256, 256, 0, s>>>(xin, bStride, CIN, xx);
  knn_kernel<<<dim3(NPT, BATCH), 128, 0, s>>>(xin, bStride, CIN, xx, idxK);
  zero_kernel<<<1, 32, 0, s>>>(gsum, 16);                  // gsum[8] + gsumsq[8]
  edgeconv_pass1<CIN, COUT><<<dim3(NPT, BATCH), 256, 0, s>>>(xin, bStride, W, idxK,
                                                             hmax, hmin, gsum, gsumsq);
  const float invcount = 1.0f/((float)(COUT/2)*(float)NPT*(float)KNN);
  ec_finalize<<<BATCH*COUT*NPT/256, 256, 0, s>>>(hmax, hmin, gsum, gsumsq,
                                                 gamma, beta, COUT, invcount,
                                                 outbase, obStride);
}

extern "C" void kernel_launch(void* const* d_in, const int* in_sizes, int n_in,
                              void* d_out, int out_size, void* d_ws, size_t ws_size,
                              hipStream_t stream){
  const float* x     = (const float*)d_in[0];
  const float* W1    = (const float*)d_in[1];
  const float* g1    = (const float*)d_in[2];
  const float* b1    = (const float*)d_in[3];
  const float* W2    = (const float*)d_in[4];
  const float* g2    = (const float*)d_in[5];
  const float* b2    = (const float*)d_in[6];
  const float* W3    = (const float*)d_in[7];
  const float* g3    = (const float*)d_in[8];
  const float* b3    = (const float*)d_in[9];
  const float* Wm    = (const float*)d_in[10];
  const float* bm    = (const float*)d_in[11];
  const float* gm    = (const float*)d_in[12];
  const float* betam = (const float*)d_in[13];

  float* out  = (float*)d_out;
  float* feat = out + BATCH*1024;                // x_features (B, 256, N)
  const long featB = 256L*NPT;

  // workspace carve-up (~11 MB)
  int*   idxK  = (int*)d_ws;                               // B*N*80 ints
  float* xx    = (float*)(idxK + (long)BATCH*NPT*KNN);     // B*N
  float* hmax  = xx   + (long)BATCH*NPT;                   // B*128*N
  float* hmin  = hmax + (long)BATCH*128*NPT;               // B*128*N
  float* gsum  = hmin + (long)BATCH*128*NPT;               // 8 (+8 gsumsq)
  float* gsq   = gsum + 8;
  float* hmax2 = gsq  + 8;                                 // B*1024
  float* hmin2 = hmax2 + BATCH*1024;                       // B*1024
  float* gs2   = hmin2 + BATCH*1024;                       // 32 (+32)
  float* gss2  = gs2 + 32;

  // layer 1: x (B,3,N) -> feat channels [0,64)
  run_edgeconv<3, 64>(x, 3L*NPT, W1, g1, b1, feat, featB,
                      idxK, xx, hmax, hmin, gsum, gsq, stream);
  // layer 2: x1 -> feat channels [64,128)
  run_edgeconv<64, 64>(feat, featB, W2, g2, b2, feat + 64L*NPT, featB,
                       idxK, xx, hmax, hmin, gsum, gsq, stream);
  // layer 3: x2 -> feat channels [128,256)
  run_edgeconv<64, 128>(feat + 64L*NPT, featB, W3, g3, b3, feat + 128L*NPT, featB,
                        idxK, xx, hmax, hmin, gsum, gsq, stream);

  // final MLP + GN(8) + ReLU + max over N
  initmm_kernel<<<16, 256, 0, stream>>>(hmax2, hmin2, BATCH*1024);
  zero_kernel<<<1, 64, 0, stream>>>(gs2, 64);              // gs2[32] + gss2[32]
  mlp_pass1<<<dim3(NPT/16, BATCH), 256, 0, stream>>>(feat, featB, Wm, bm,
                                                     hmax2, hmin2, gs2, gss2);
  mlp_finalize<<<16, 256, 0, stream>>>(hmax2, hmin2, gs2, gss2, gm, betam, out);
}